// VPLayer_71373766525316
// MI455X (gfx1250) — compile-verified
//
#include <hip/hip_runtime.h>
#include <math.h>

// CDNA5 / gfx1250. wave32. Matrix ops via V_WMMA_F32_16X16X4_F32 (full fp32).
typedef __attribute__((ext_vector_type(2))) float v2f;
typedef __attribute__((ext_vector_type(8))) float v8f;

#define VP_B 8
#define VP_S 16384
#define VP_D 512
#define VP_ROWS 192          // 2*32 + 2*64 output rows per batch
#define VP_FACTOR 10.0f

// One workgroup per (b, k): 256 threads = 8 waves; wave w owns d in [64w, 64w+64)
// as four 16-wide WMMA N-tiles. Accumulates sx = sum_s w(s)*x[b,s,d] and
// sx2 = sum_s w(s)*x^2 on the matrix pipe, then writes mean / std.
__global__ __launch_bounds__(256) void vp_seg_kernel(
    const float* __restrict__ x, const float* __restrict__ scores,
    float* __restrict__ out, int K, int mean_off, int var_off)
{
  const int b = blockIdx.x / K;
  const int k = blockIdx.x % K;

  // Segment params: softmax(scores*F) -> cumsum -> [p,q), denom. K <= 64: trivial.
  __shared__ float sh[3];
  if (threadIdx.x == 0) {
    float m = -3.402823466e38f;
    for (int i = 0; i < K; ++i) m = fmaxf(m, scores[i] * VP_FACTOR);
    float sum = 0.f;
    for (int i = 0; i < K; ++i) sum += expf(scores[i] * VP_FACTOR - m);
    const float inv = 1.f / sum;
    float c = 0.f, pv = 0.f;
    for (int i = 0; i <= k; ++i) { pv = c; c += expf(scores[i] * VP_FACTOR - m) * inv; }
    const float p = pv * (float)VP_S;
    const float q = fminf(c * (float)VP_S, (float)VP_S - 0.01f);
    sh[0] = p; sh[1] = q; sh[2] = q - p;
  }
  __syncthreads();
  const float p = sh[0], q = sh[1], den = sh[2];

  const int lane  = threadIdx.x & 31;
  const int wave  = threadIdx.x >> 5;
  const int half  = lane >> 4;     // lane half selects K={0,1} vs K={2,3} slots
  const int lc    = lane & 15;     // N / M position within tile
  const int d_base = wave * 64;

  const float* __restrict__ xb = x + (size_t)b * VP_S * VP_D;

  v8f acc1[4] = {}, acc2[4] = {};  // fp32 16x16 accumulators (only row 0 used)

  int s_begin = (int)floorf(p); if (s_begin < 0) s_begin = 0;
  int s_end   = (int)ceilf(q);  if (s_end > VP_S) s_end = VP_S;

  for (int s0 = (s_begin & ~3); s0 < s_end; s0 += 4) {
    const int r0 = s0 + half * 2;      // this lane-half supplies rows r0, r0+1
    const int r1 = r0 + 1;
    // Fractional overlap weights; 0 outside [p,q] so alignment over-read is harmless.
    v2f a;
    a.x = fmaxf(fminf(q, (float)(r0 + 1)) - fmaxf(p, (float)r0), 0.f);
    a.y = fmaxf(fminf(q, (float)(r1 + 1)) - fmaxf(p, (float)r1), 0.f);
    const float* row0 = xb + (size_t)min(r0, VP_S - 1) * VP_D + d_base;
    const float* row1 = xb + (size_t)min(r1, VP_S - 1) * VP_D + d_base;
    __builtin_prefetch(row0 + 4 * VP_D, 0, 1);   // global_prefetch_b8: next 4-row group
    __builtin_prefetch(row1 + 4 * VP_D, 0, 1);
#pragma unroll
    for (int j = 0; j < 4; ++j) {
      const int col = j * 16 + lc;               // 64B coalesced per lane-half
      v2f bv, bsq;
      bv.x = row0[col];
      bv.y = row1[col];
      bsq.x = bv.x * bv.x;
      bsq.y = bv.y * bv.y;
      // D = A(16x4 broadcast weights) x B(4x16 x-tile) + C ; all D rows identical.
      acc1[j] = __builtin_amdgcn_wmma_f32_16x16x4_f32(
          false, a, false, bv,  (short)0, acc1[j], false, false);
      acc2[j] = __builtin_amdgcn_wmma_f32_16x16x4_f32(
          false, a, false, bsq, (short)0, acc2[j], false, false);
    }
  }

  // Row 0 of D lives in D-VGPR0 (element 0) for lanes 0..15 (N = lc).
  if (half == 0) {
#pragma unroll
    for (int j = 0; j < 4; ++j) {
      const int d    = d_base + j * 16 + lc;
      const float sx  = acc1[j][0];
      const float sx2 = acc2[j][0];
      const float mean = sx / den;
      const float var  = sqrtf(fmaxf(sx2 - mean * mean * den, 0.f) / den);
      out[((size_t)b * VP_ROWS + mean_off + k) * VP_D + d] = mean;
      out[((size_t)b * VP_ROWS + var_off  + k) * VP_D + d] = var;
    }
  }
}

extern "C" void kernel_launch(void* const* d_in, const int* in_sizes, int n_in,
                              void* d_out, int out_size, void* d_ws, size_t ws_size,
                              hipStream_t stream) {
  (void)in_sizes; (void)n_in; (void)out_size; (void)d_ws; (void)ws_size;
  const float* x  = (const float*)d_in[0];
  const float* s0 = (const float*)d_in[1];   // 32 scores
  const float* s1 = (const float*)d_in[2];   // 64 scores
  float* out = (float*)d_out;
  // Output rows per batch: [mean0 0..31 | var0 32..63 | mean1 64..127 | var1 128..191]
  vp_seg_kernel<<<VP_B * 32, 256, 0, stream>>>(x, s0, out, 32, 0, 32);
  vp_seg_kernel<<<VP_B * 64, 256, 0, stream>>>(x, s1, out, 64, 64, 128);
}